// NWJ_26680336843004
// MI455X (gfx1250) — compile-verified
//
#include <hip/hip_runtime.h>
#include <math.h>

#define N_S   1024
#define XD    128
#define HID   256
#define TI    64
#define TJ    64
#define KC    64

typedef float v2f __attribute__((ext_vector_type(2)));
typedef float v8f __attribute__((ext_vector_type(8)));

// outT[m][n] = sum_{k=0..127} W1[kbase+k][m] * S[n][k]  (+ b1[m] if addBias)
// One 16x16 f32 tile per wave via V_WMMA_F32_16X16X4_F32 (K-loop of 32).
// A-frag (16x4 f32, ISA 7.12.2): lane L: m=L&15, khalf=L>>4; v0=A[m][k0+2*kh], v1=A[m][k0+2*kh+1]
// B-frag (4x16 f32, symmetric):  lane L: n=L&15;            v0=B[k0+2*kh][n], v1=B[k0+2*kh+1][n]
// C/D:   VGPR v: lanes0-15 -> M=v, lanes16-31 -> M=v+8; N=lane&15
__global__ void nwj_wmma_gemm_xT(const float* __restrict__ W1,
                                 const float* __restrict__ S,
                                 const float* __restrict__ b1,
                                 int kbase, int addBias,
                                 float* __restrict__ outT) {
  const int lane = threadIdx.x & 31;
  const int wave = threadIdx.x >> 5;
  const int tile = blockIdx.x * 8 + wave;     // 1024 tiles: 16 (m) x 64 (n)
  const int m0 = (tile >> 6) << 4;            // hid-row base
  const int n0 = (tile & 63) << 4;            // sample-col base
  const int ml = lane & 15;
  const int kh = (lane >> 4) << 1;            // 0 or 2

  v8f c = {};
  for (int k0 = 0; k0 < XD; k0 += 4) {
    v2f a, b;
    const int ka = kbase + k0 + kh;
    a.x = W1[(ka + 0) * HID + m0 + ml];
    a.y = W1[(ka + 1) * HID + m0 + ml];
    const float* sp = S + (n0 + ml) * XD + (k0 + kh);
    b.x = sp[0];
    b.y = sp[1];
    c = __builtin_amdgcn_wmma_f32_16x16x4_f32(false, a, false, b,
                                              (short)0, c, false, false);
  }

  const int mh = (lane >> 4) << 3;            // 0 or 8
  #pragma unroll
  for (int v = 0; v < 8; ++v) {
    const int m = m0 + mh + v;
    float val = c[v];
    if (addBias) val += b1[m];
    outT[m * N_S + n0 + ml] = val;
  }
}

// T1[i,j] = sum_k relu(xpT[k][j] + ypbT[k][i]) * W2[k] + b2 - 1
// Each block: 64i x 64j tile; emits per-(i, jblock) partial (max, sumexp).
__global__ void nwj_pairwise_lse(const float* __restrict__ xpT,
                                 const float* __restrict__ ypbT,
                                 const float* __restrict__ W2,
                                 const float* __restrict__ b2p,
                                 float* __restrict__ pmax,
                                 float* __restrict__ psum) {
  __shared__ float xs[KC][TJ];
  __shared__ float ys[KC][TI];
  __shared__ float w2s[KC];

  const int tid = threadIdx.x;
  const int jb = blockIdx.x;                  // 0..15
  const int ib = blockIdx.y;                  // 0..15
  const int j0 = jb * TJ;
  const int i0 = ib * TI;
  const int tx = tid & 15;                    // j-group
  const int ty = tid >> 4;                    // i-group

  float acc[4][4];
  #pragma unroll
  for (int ii = 0; ii < 4; ++ii)
    #pragma unroll
    for (int jj = 0; jj < 4; ++jj) acc[ii][jj] = 0.0f;

  for (int kc = 0; kc < HID; kc += KC) {
    #pragma unroll
    for (int q = 0; q < 16; ++q) {            // 4096 elems / 256 threads
      const int e = q * 256 + tid;
      const int kk = e >> 6;
      const int col = e & 63;
      xs[kk][col] = xpT[(kc + kk) * N_S + j0 + col];
      ys[kk][col] = ypbT[(kc + kk) * N_S + i0 + col];
    }
    if (tid < KC) w2s[tid] = W2[kc + tid];
    __syncthreads();

    for (int kk = 0; kk < KC; ++kk) {
      const float4 av = *reinterpret_cast<const float4*>(&xs[kk][tx * 4]);
      const float4 bv = *reinterpret_cast<const float4*>(&ys[kk][ty * 4]);
      const float w = w2s[kk];
      const float aj[4] = {av.x, av.y, av.z, av.w};
      const float bi[4] = {bv.x, bv.y, bv.z, bv.w};
      #pragma unroll
      for (int ii = 0; ii < 4; ++ii)
        #pragma unroll
        for (int jj = 0; jj < 4; ++jj)
          acc[ii][jj] = fmaf(fmaxf(aj[jj] + bi[ii], 0.0f), w, acc[ii][jj]);
    }
    __syncthreads();
  }

  const float base = b2p[0] - 1.0f;
  #pragma unroll
  for (int ii = 0; ii < 4; ++ii) {
    float t[4];
    float m = -INFINITY;
    #pragma unroll
    for (int jj = 0; jj < 4; ++jj) {
      t[jj] = acc[ii][jj] + base;
      m = fmaxf(m, t[jj]);
    }
    // butterfly max/sum across the 16 tx lanes (stays within half-wave)
    for (int d = 1; d < 16; d <<= 1) m = fmaxf(m, __shfl_xor(m, d, 32));
    float s = 0.0f;
    #pragma unroll
    for (int jj = 0; jj < 4; ++jj) s += __expf(t[jj] - m);
    for (int d = 1; d < 16; d <<= 1) s += __shfl_xor(s, d, 32);
    if (tx == 0) {
      const int i = i0 + ty * 4 + ii;
      pmax[i * 16 + jb] = m;
      psum[i * 16 + jb] = s;
    }
  }
}

// Per-row: combine 16 partials -> lse; val = exp(lse - log N). Also T0[i].
__global__ void nwj_finalize_rows(const float* __restrict__ xpT,
                                  const float* __restrict__ ypbT,
                                  const float* __restrict__ W2,
                                  const float* __restrict__ b2p,
                                  const float* __restrict__ pmax,
                                  const float* __restrict__ psum,
                                  float* __restrict__ t0row,
                                  float* __restrict__ vrow) {
  const int i = blockIdx.x * 256 + threadIdx.x;   // 0..1023
  float t0 = 0.0f;
  for (int k = 0; k < HID; ++k) {
    const float h = xpT[k * N_S + i] + ypbT[k * N_S + i]; // b1 folded into ypbT
    t0 = fmaf(fmaxf(h, 0.0f), W2[k], t0);
  }
  t0row[i] = t0 + b2p[0];

  float M = -INFINITY;
  #pragma unroll
  for (int jb = 0; jb < 16; ++jb) M = fmaxf(M, pmax[i * 16 + jb]);
  float S = 0.0f;
  #pragma unroll
  for (int jb = 0; jb < 16; ++jb)
    S += psum[i * 16 + jb] * __expf(pmax[i * 16 + jb] - M);
  const float lse = M + __logf(S);
  vrow[i] = __expf(lse - __logf((float)N_S));
}

__global__ void nwj_reduce_final(const float* __restrict__ t0row,
                                 const float* __restrict__ vrow,
                                 float* __restrict__ out) {
  __shared__ float st[256];
  __shared__ float sv[256];
  const int t = threadIdx.x;
  float a = 0.0f, b = 0.0f;
  for (int i = t; i < N_S; i += 256) { a += t0row[i]; b += vrow[i]; }
  st[t] = a; sv[t] = b;
  __syncthreads();
  for (int s = 128; s > 0; s >>= 1) {
    if (t < s) { st[t] += st[t + s]; sv[t] += sv[t + s]; }
    __syncthreads();
  }
  if (t == 0) out[0] = st[0] / (float)N_S - sv[0] / (float)N_S;
}

extern "C" void kernel_launch(void* const* d_in, const int* in_sizes, int n_in,
                              void* d_out, int out_size, void* d_ws, size_t ws_size,
                              hipStream_t stream) {
  const float* x  = (const float*)d_in[0];   // 1024 x 128
  const float* y  = (const float*)d_in[1];   // 1024 x 128
  const float* W1 = (const float*)d_in[2];   // 256 x 256
  const float* b1 = (const float*)d_in[3];   // 256
  const float* W2 = (const float*)d_in[4];   // 256 x 1
  const float* b2 = (const float*)d_in[5];   // 1
  float* out = (float*)d_out;

  float* ws    = (float*)d_ws;
  float* xpT   = ws;                         // 256*1024
  float* ypbT  = xpT + HID * N_S;            // 256*1024
  float* pmax  = ypbT + HID * N_S;           // 1024*16
  float* psum  = pmax + N_S * 16;            // 1024*16
  float* t0row = psum + N_S * 16;            // 1024
  float* vrow  = t0row + N_S;                // 1024

  // xpT = Wx^T @ x^T ; ypbT = Wy^T @ y^T + b1
  nwj_wmma_gemm_xT<<<128, 256, 0, stream>>>(W1, x, b1, 0,   0, xpT);
  nwj_wmma_gemm_xT<<<128, 256, 0, stream>>>(W1, y, b1, XD,  1, ypbT);

  nwj_pairwise_lse<<<dim3(16, 16), 256, 0, stream>>>(xpT, ypbT, W2, b2, pmax, psum);
  nwj_finalize_rows<<<4, 256, 0, stream>>>(xpT, ypbT, W2, b2, pmax, psum, t0row, vrow);
  nwj_reduce_final<<<1, 256, 0, stream>>>(t0row, vrow, out);
}